// CovarianceResidualError_22574348108101
// MI455X (gfx1250) — compile-verified
//
#include <hip/hip_runtime.h>
#include <hip/hip_bf16.h>

typedef __attribute__((ext_vector_type(2))) float v2f;
typedef __attribute__((ext_vector_type(8))) float v8f;

#define N_ROWS 131072
#define D_DIM  128
#define O_DIM  256

#define WAVES_PER_TILE 128                       // row-groups per 16-column tile
#define TOTAL_WAVES    (16 * WAVES_PER_TILE)     // 2048 waves
#define ROWS_PER_WAVE  (N_ROWS / WAVES_PER_TILE) // 1024
#define SG_BLOCKS      512
// ws float layout: [0,512) = Sg block partials; [512, 512+TOTAL_WAVES*32) = wave partials
// (per wave: 16 floats C1, 16 floats S). Total ~258 KB.

// ---------- Kernel 1: per-block partial sums of graph_emb[:,0] ----------
__global__ __launch_bounds__(256) void sg_partial_kernel(const float* __restrict__ g,
                                                         float* __restrict__ ws) {
  __shared__ float red[256];
  int i = blockIdx.x * 256 + threadIdx.x;
  red[threadIdx.x] = g[(size_t)i * D_DIM];  // column 0, stride D_DIM
  __syncthreads();
  for (int s = 128; s > 0; s >>= 1) {
    if (threadIdx.x < s) red[threadIdx.x] += red[threadIdx.x + s];
    __syncthreads();
  }
  if (threadIdx.x == 0) ws[blockIdx.x] = red[0];
}

// ---------- Kernel 2: WMMA rank-2 reduction over errors ----------
// D = A(16x4) x B(4x16) + C, accumulated over all row chunks.
// A row0 = g0 weights, A row1 = ones  =>  D[0][n] = C1 partial, D[1][n] = S partial.
// f32 operand layout (wave32): A: lane=M, VGPR0 holds K=0 (lanes 0-15) / K=2 (lanes 16-31),
// VGPR1 holds K=1 / K=3. B mirrors with lane=N. C/D: VGPR r = row M=r (lanes 0-15).
__global__ __launch_bounds__(256) void cov_wmma_kernel(const float* __restrict__ errors,
                                                       const float* __restrict__ g,
                                                       float* __restrict__ ws) {
  const int lane    = threadIdx.x & 31;
  const int wave    = blockIdx.x * 8 + (threadIdx.x >> 5);
  const int colTile = wave & 15;             // 16 tiles of 16 columns
  const int rowGrp  = wave >> 4;             // 128 row groups
  const int j0      = colTile * 16;
  const int rowBase = rowGrp * ROWS_PER_WAVE;
  const int rowEnd  = rowBase + ROWS_PER_WAVE;

  const int m      = lane & 15;              // matrix row / column index within tile
  const int rowOff = (lane < 16) ? 0 : 2;    // K half select for lanes 16-31
  const float* eb  = errors + (size_t)(j0 + m) + (size_t)rowOff * O_DIM;

  v8f acc = {};
  for (int i0 = rowBase; i0 < rowEnd; i0 += 32) {
    // coalesced-ish gather of 32 g0 values (stride D_DIM; lines are L2-hot from kernel 1)
    float gch = g[(size_t)(i0 + lane) * D_DIM];

    // stage all 16 error loads for this 32-row block (16 cols x 32 rows = 2 KB/wave)
    float bx[8], by[8];
#pragma unroll
    for (int c = 0; c < 8; ++c) {
      size_t base = (size_t)(i0 + 4 * c) * O_DIM;
      bx[c] = eb[base];          // rows r0+rowOff
      by[c] = eb[base + O_DIM];  // rows r0+rowOff+1
    }

    // uniform prefetch of the next row block (lowers to global_prefetch_b8)
    if (i0 + 32 < rowEnd) {
      __builtin_prefetch(&eb[(size_t)(i0 + 32) * O_DIM], 0, 1);
    }

#pragma unroll
    for (int c = 0; c < 8; ++c) {
      int   kx = 4 * c + rowOff;
      float gx = __shfl(gch, kx, 32);
      float gy = __shfl(gch, kx + 1, 32);
      v2f a, b;
      a.x = (m == 0) ? gx : ((m == 1) ? 1.0f : 0.0f);
      a.y = (m == 0) ? gy : ((m == 1) ? 1.0f : 0.0f);
      b.x = bx[c];
      b.y = by[c];
      acc = __builtin_amdgcn_wmma_f32_16x16x4_f32(
          /*neg_a=*/false, a, /*neg_b=*/false, b,
          /*c_mod=*/(short)0, acc, /*reuse_a=*/false, /*reuse_b=*/false);
    }
  }

  // D row M=0 (C1) and M=1 (S) live in acc[0]/acc[1] of lanes 0-15
  if (lane < 16) {
    float* wp = ws + 512 + (size_t)wave * 32;
    wp[lane]      = acc[0];
    wp[16 + lane] = acc[1];
  }
}

// ---------- Kernel 3: deterministic final reduction ----------
__global__ __launch_bounds__(256) void finalize_kernel(const float* __restrict__ ws,
                                                       float* __restrict__ out) {
  __shared__ float red[256];
  __shared__ float sgSh;
  int t = threadIdx.x;

  // reduce Sg over 512 block partials (fixed order)
  red[t] = ws[t] + ws[t + 256];
  __syncthreads();
  for (int s = 128; s > 0; s >>= 1) {
    if (t < s) red[t] += red[t + s];
    __syncthreads();
  }
  if (t == 0) sgSh = red[0];
  __syncthreads();
  float Sg = sgSh;

  // one thread per output column: fold 128 wave partials in fixed order
  int colTile = t >> 4, jj = t & 15;
  float c1 = 0.0f, sj = 0.0f;
  for (int rg = 0; rg < WAVES_PER_TILE; ++rg) {
    const float* wp = ws + 512 + (size_t)(rg * 16 + colTile) * 32;
    c1 += wp[jj];
    sj += wp[16 + jj];
  }
  float col = c1 - Sg * sj * (1.0f / (float)N_ROWS);

  red[t] = fabsf(col);
  __syncthreads();
  for (int s = 128; s > 0; s >>= 1) {
    if (t < s) red[t] += red[t + s];
    __syncthreads();
  }
  if (t == 0) out[0] = -red[0];
}

extern "C" void kernel_launch(void* const* d_in, const int* in_sizes, int n_in,
                              void* d_out, int out_size, void* d_ws, size_t ws_size,
                              hipStream_t stream) {
  // inputs: 0=targets 1=out0 2=out1 3=graph_emb(N,128) 4=errors(N,256)
  const float* graph_emb = (const float*)d_in[3];
  const float* errors    = (const float*)d_in[4];
  float* ws  = (float*)d_ws;
  float* out = (float*)d_out;

  sg_partial_kernel<<<SG_BLOCKS, 256, 0, stream>>>(graph_emb, ws);
  cov_wmma_kernel<<<TOTAL_WAVES / 8, 256, 0, stream>>>(errors, graph_emb, ws);
  finalize_kernel<<<1, 256, 0, stream>>>(ws, out);
}